// HyperNet_89395449299629
// MI455X (gfx1250) — compile-verified
//
#include <hip/hip_runtime.h>
#include <hip/hip_bf16.h>

// ---------------------------------------------------------------------------
// HyperNet for MI455X (gfx1250, wave32).
//   Stage 1 (tiny): h2[h] = relu(W2[h] @ relu(W1[h] @ cam + b1[h]) + b2[h])
//                   -> padded H[16][256] (rows 10..15 zero) in workspace.
//   Stage 2 (204 MB stream): per 16-row tile of Wf, D = Wf_tile x H^T via
//                   V_WMMA_F32_16X16X4_F32, then out[t] = D[m, head_idx[t]] + bf[t].
// Memory bound: 204 MB / 23.3 TB/s ~ 8.7us; WMMA demand ~186 TF/s f32 (< ceiling).
// ---------------------------------------------------------------------------

typedef __attribute__((ext_vector_type(2))) float v2f;
typedef __attribute__((ext_vector_type(8))) float v8f;

#define N_HEADS 10
#define HID     256
#define TOTAL_T 198915
#define TILES   ((TOTAL_T + 15) / 16)   // 12433
#define WAVES_PER_BLOCK 8

// ---------------------------------------------------------------------------
// Stage 1: compute hidden states. 16 blocks (one per padded head slot), 256 thr.
// HT layout in workspace: HT[n * 256 + k] = h2[n][k]  (n = head slot, k = hidden)
// ---------------------------------------------------------------------------
__global__ void __launch_bounds__(256)
hyper_hidden_kernel(const float* __restrict__ cam,
                    const float* __restrict__ W1, const float* __restrict__ b1,
                    const float* __restrict__ W2, const float* __restrict__ b2,
                    float* __restrict__ HT) {
  __shared__ float h1[HID];
  const int h = blockIdx.x;
  const int o = threadIdx.x;

  if (h >= N_HEADS) {               // pad rows 10..15 with zeros
    HT[h * HID + o] = 0.0f;
    return;
  }

  // Layer 1: 256x3 matvec + bias + relu
  const float c0 = cam[0], c1 = cam[1], c2 = cam[2];
  const float* w1r = W1 + (size_t)(h * HID + o) * 3;
  float s1 = b1[h * HID + o] + w1r[0] * c0 + w1r[1] * c1 + w1r[2] * c2;
  h1[o] = fmaxf(s1, 0.0f);
  __syncthreads();

  // Layer 2: 256x256 matvec + bias + relu (tiny: 2.6 MB of W2 total)
  const float4* w2r = (const float4*)(W2 + (size_t)h * HID * HID + (size_t)o * HID);
  const float4* h1v = (const float4*)h1;
  float acc = b2[h * HID + o];
#pragma unroll 4
  for (int i = 0; i < HID / 4; ++i) {
    float4 w = w2r[i];
    float4 x = h1v[i];
    acc += w.x * x.x + w.y * x.y + w.z * x.z + w.w * x.w;
  }
  HT[h * HID + o] = fmaxf(acc, 0.0f);
}

// ---------------------------------------------------------------------------
// Stage 2: stream Wf. One wave32 per 16-row tile; 8 waves / block.
// A (Wf tile): lane<16 -> row=lane, K={k0,k0+1}; lane>=16 -> K={k0+2,k0+3}.
// B (H):       lane     -> col=lane&15, same K pair, from LDS (ds_load_b64).
// D:           VGPR v of lane L holds (M = v + 8*(L>=16), N = L&15).
// ---------------------------------------------------------------------------
__global__ void __launch_bounds__(256)
hyper_final_kernel(const float* __restrict__ Wf,
                   const float* __restrict__ bf,
                   const int*   __restrict__ head_idx,
                   const float* __restrict__ HTg,
                   float* __restrict__ out) {
  __shared__ float HT[16 * HID];      // 16 KB, resident for whole block

  const int tid = threadIdx.x;
  for (int i = tid; i < 16 * HID; i += 256) HT[i] = HTg[i];
  __syncthreads();

  const int wave = tid >> 5;
  const int lane = tid & 31;
  const int tile = blockIdx.x * WAVES_PER_BLOCK + wave;
  if (tile >= TILES) return;          // wave-uniform: EXEC stays all-ones

  const int t0 = tile * 16;
  const int m  = lane & 15;           // tile row for A / column for B
  int rowA = t0 + m;
  if (rowA > TOTAL_T - 1) rowA = TOTAL_T - 1;   // clamp tail rows (loads stay in-bounds)
  const int kphase = (lane >> 4) * 2; // 0 for lanes 0..15, 2 for lanes 16..31

  const float* Ap = Wf + (size_t)rowA * HID + kphase;
  const float* Bp = &HT[m * HID + kphase];

  v8f acc = {0.f, 0.f, 0.f, 0.f, 0.f, 0.f, 0.f, 0.f};

#pragma unroll 8
  for (int k0 = 0; k0 < HID; k0 += 4) {
    v2f a = *(const v2f*)(Ap + k0);   // global_load_b64: K = k0+kphase, +1
    v2f b = *(const v2f*)(Bp + k0);   // ds_load_b64:     same K pair, col m
    // 8 args: (neg_a, A, neg_b, B, c_mod, C, reuse_a, reuse_b)
    acc = __builtin_amdgcn_wmma_f32_16x16x4_f32(
        false, a, false, b, (short)0, acc, false, false);
  }

  // Extract D[m, head_idx[t0+m]] for rows m = 0..15 (held by lanes 0..15).
  int t = t0 + m;
  int tcl = (t < TOTAL_T) ? t : (TOTAL_T - 1);
  const int hsel = head_idx[tcl];                  // 0..9
  const int sel  = hsel + ((lane & 8) ? 16 : 0);   // source lane: N + 16*(M>=8)
  float res = 0.0f;
#pragma unroll
  for (int v = 0; v < 8; ++v) {                    // M = v (+8): value in VGPR v=M&7
    float tv = __shfl(acc[v], sel, 32);
    if ((lane & 7) == v) res = tv;
  }

  if (lane < 16 && t < TOTAL_T) out[t] = res + bf[t];
}

// ---------------------------------------------------------------------------
extern "C" void kernel_launch(void* const* d_in, const int* in_sizes, int n_in,
                              void* d_out, int out_size, void* d_ws, size_t ws_size,
                              hipStream_t stream) {
  const float* cam  = (const float*)d_in[0];
  const float* W1   = (const float*)d_in[1];
  const float* b1   = (const float*)d_in[2];
  const float* W2   = (const float*)d_in[3];
  const float* b2   = (const float*)d_in[4];
  const float* Wf   = (const float*)d_in[5];
  const float* bf   = (const float*)d_in[6];
  const int*   hidx = (const int*)d_in[7];
  float* out = (float*)d_out;
  float* HTg = (float*)d_ws;          // 16*256 floats = 16 KB scratch

  hyper_hidden_kernel<<<16, 256, 0, stream>>>(cam, W1, b1, W2, b2, HTg);

  const int blocks = (TILES + WAVES_PER_BLOCK - 1) / WAVES_PER_BLOCK;  // 1555
  hyper_final_kernel<<<blocks, 256, 0, stream>>>(Wf, bf, hidx, HTg, out);
}